// ResidualGCN_4904852652788
// MI455X (gfx1250) — compile-verified
//
#include <hip/hip_runtime.h>
#include <hip/hip_bf16.h>
#include <stdint.h>

typedef float v2f __attribute__((ext_vector_type(2)));
typedef float v4f __attribute__((ext_vector_type(4)));
typedef float v8f __attribute__((ext_vector_type(8)));

#define DFEAT 256
#define BN_EPS 1e-5f

// ---------------- degree / normalization ----------------
__global__ void deg_kernel(const long long* __restrict__ col,
                           const float* __restrict__ ew,
                           float* __restrict__ deg, int E) {
    int e = blockIdx.x * blockDim.x + threadIdx.x;
    if (e < E) atomicAdd(&deg[(int)col[e]], ew[e]);
}

__global__ void dis_kernel(const float* __restrict__ deg,
                           float* __restrict__ dis, int N) {
    int i = blockIdx.x * blockDim.x + threadIdx.x;
    if (i < N) dis[i] = rsqrtf(deg[i] + 1.0f);   // +1 = self loop weight
}

// ---------------- fp32 WMMA GEMM: C[M x 256] = A[M x 256] * B[256 x 256] ----------------
// block = 256 threads = 8 waves; each wave owns a 16x16 tile; grid = (ceil(M/128), 16)
__global__ __launch_bounds__(256) void gemm_wmma_f32(const float* __restrict__ A,
                                                     const float* __restrict__ B,
                                                     float* __restrict__ C, int M) {
    const int wave = threadIdx.x >> 5;
    const int lane = threadIdx.x & 31;
    const int mBase = blockIdx.x * 128 + wave * 16;
    if (mBase >= M) return;                       // wave-uniform exit (EXEC stays all-1s)
    const int nBase = blockIdx.y * 16;
    const int m    = lane & 15;                   // also serves as n for B/C fragments
    const int half = lane >> 4;

    v8f c = {};
    const float* arow = A + (size_t)(mBase + m) * DFEAT;
    #pragma unroll 4
    for (int k0 = 0; k0 < DFEAT; k0 += 4) {
        const int k = k0 + 2 * half;
        v2f a, b;
        a.x = arow[k];
        a.y = arow[k + 1];
        b.x = B[(size_t)k       * DFEAT + nBase + m];
        b.y = B[(size_t)(k + 1) * DFEAT + nBase + m];
        c = __builtin_amdgcn_wmma_f32_16x16x4_f32(false, a, false, b,
                                                  (short)0, c, false, false);
    }
    float* crow = C + (size_t)(mBase + 8 * half) * DFEAT + nBase + m;
    #pragma unroll
    for (int r = 0; r < 8; ++r) crow[(size_t)r * DFEAT] = c[r];
}

// ---------------- z[i] = dis[i]^2 * h[i]  (self-loop term initializes agg) ----------------
__global__ __launch_bounds__(256) void selfloop_kernel(const float* __restrict__ h,
                                                       const float* __restrict__ dis,
                                                       float* __restrict__ z, int N) {
    size_t idx = (size_t)blockIdx.x * blockDim.x + threadIdx.x;   // float4 units
    size_t total = (size_t)N * (DFEAT / 4);
    if (idx >= total) return;
    int i = (int)(idx / (DFEAT / 4));
    float s = dis[i];
    s = s * s;
    v4f v = ((const v4f*)h)[idx];
    ((v4f*)z)[idx] = v * s;
}

// ---------------- edge scatter: z[col] += dis[row]*ew*dis[col] * h[row] ----------------
// one wave per edge, lane handles 8 features (2x float4 gather, 8 f32 atomics)
__global__ __launch_bounds__(256) void scatter_kernel(const long long* __restrict__ row,
                                                      const long long* __restrict__ col,
                                                      const float* __restrict__ ew,
                                                      const float* __restrict__ dis,
                                                      const float* __restrict__ h,
                                                      float* __restrict__ z, int E) {
    int e = blockIdx.x * 8 + (threadIdx.x >> 5);
    if (e >= E) return;
    const int lane = threadIdx.x & 31;
    const int r = (int)row[e];
    const int t = (int)col[e];
    const float coef = dis[r] * ew[e] * dis[t];
    const float* hp = h + (size_t)r * DFEAT + lane * 8;
    float* zp = z + (size_t)t * DFEAT + lane * 8;
    __builtin_prefetch(zp, 1, 0);                 // global_prefetch_b8 on the scatter target
    v4f v0 = *(const v4f*)hp;
    v4f v1 = *(const v4f*)(hp + 4);
    v0 *= coef;
    v1 *= coef;
    atomicAdd(zp + 0, v0.x); atomicAdd(zp + 1, v0.y);
    atomicAdd(zp + 2, v0.z); atomicAdd(zp + 3, v0.w);
    atomicAdd(zp + 4, v1.x); atomicAdd(zp + 5, v1.y);
    atomicAdd(zp + 6, v1.z); atomicAdd(zp + 7, v1.w);
}

// ---------------- BN column stats: sum + sumsq partials ----------------
__global__ __launch_bounds__(256) void stats_kernel(const float* __restrict__ z,
                                                    float* __restrict__ stats, int N) {
    const int t = threadIdx.x;                    // column
    int r0 = blockIdx.x * 256;
    int r1 = r0 + 256; if (r1 > N) r1 = N;
    float s = 0.f, sq = 0.f;
    for (int r = r0; r < r1; ++r) {
        float v = z[(size_t)r * DFEAT + t];
        s += v; sq += v * v;
    }
    atomicAdd(&stats[t], s);
    atomicAdd(&stats[DFEAT + t], sq);
}

// ---------------- BN finalize: scale/shift (bias cancels in mean subtraction) ----------------
__global__ void bnfin_kernel(const float* __restrict__ stats,
                             const float* __restrict__ g,
                             const float* __restrict__ be,
                             float* __restrict__ bn, int N) {
    const int t = threadIdx.x;
    const float invN = 1.0f / (float)N;
    float mean = stats[t] * invN;
    float var  = stats[DFEAT + t] * invN - mean * mean;
    float scale = g[t] * rsqrtf(var + BN_EPS);
    bn[t] = scale;
    bn[DFEAT + t] = be[t] - mean * scale;
}

// ---------------- y = relu(scale*z + shift) ----------------
__global__ __launch_bounds__(256) void apply_relu_kernel(const float* __restrict__ z,
                                                         const float* __restrict__ bn,
                                                         float* __restrict__ y, int N) {
    size_t idx = (size_t)blockIdx.x * blockDim.x + threadIdx.x;   // float4 units
    size_t total = (size_t)N * (DFEAT / 4);
    if (idx >= total) return;
    int c4 = ((int)(idx & 63)) * 4;
    v4f v  = ((const v4f*)z)[idx];
    v4f sc = *(const v4f*)(bn + c4);
    v4f sh = *(const v4f*)(bn + DFEAT + c4);
    v = v * sc + sh;
    v.x = fmaxf(v.x, 0.f); v.y = fmaxf(v.y, 0.f);
    v.z = fmaxf(v.z, 0.f); v.w = fmaxf(v.w, 0.f);
    ((v4f*)y)[idx] = v;
}

// ---------------- out = relu(scale*z + shift + x)  (in-place on z) ----------------
__global__ __launch_bounds__(256) void apply_final_kernel(float* __restrict__ z,
                                                          const float* __restrict__ bn,
                                                          const float* __restrict__ x, int N) {
    size_t idx = (size_t)blockIdx.x * blockDim.x + threadIdx.x;
    size_t total = (size_t)N * (DFEAT / 4);
    if (idx >= total) return;
    int c4 = ((int)(idx & 63)) * 4;
    v4f v  = ((const v4f*)z)[idx];
    v4f xv = ((const v4f*)x)[idx];
    v4f sc = *(const v4f*)(bn + c4);
    v4f sh = *(const v4f*)(bn + DFEAT + c4);
    v = v * sc + sh + xv;
    v.x = fmaxf(v.x, 0.f); v.y = fmaxf(v.y, 0.f);
    v.z = fmaxf(v.z, 0.f); v.w = fmaxf(v.w, 0.f);
    ((v4f*)z)[idx] = v;
}

extern "C" void kernel_launch(void* const* d_in, const int* in_sizes, int n_in,
                              void* d_out, int out_size, void* d_ws, size_t ws_size,
                              hipStream_t stream) {
    const float*     x   = (const float*)d_in[0];
    const long long* ei  = (const long long*)d_in[1];     // int64 [2, E]
    const float*     ew  = (const float*)d_in[2];
    const float*     W1  = (const float*)d_in[3];
    // d_in[4] = b1: cancels in BN (and is zero)
    const float*     g1  = (const float*)d_in[5];
    const float*     be1 = (const float*)d_in[6];
    const float*     W2  = (const float*)d_in[7];
    // d_in[8] = b2: cancels in BN
    const float*     g2  = (const float*)d_in[9];
    const float*     be2 = (const float*)d_in[10];

    const int N = in_sizes[0] / DFEAT;
    const int E = in_sizes[2];
    const long long* rowp = ei;
    const long long* colp = ei + E;

    // workspace layout (floats)
    float* big   = (float*)d_ws;                  // N*DFEAT   (ping-pong buffer)
    float* deg   = big + (size_t)N * DFEAT;       // N
    float* dis   = deg + N;                       // N
    float* stats = dis + N;                       // 2*DFEAT
    float* bn    = stats + 2 * DFEAT;             // 2*DFEAT
    float* out   = (float*)d_out;                 // N*DFEAT

    const int TB = 256;
    const int gE    = (E + TB - 1) / TB;
    const int gN    = (N + TB - 1) / TB;
    const int gVec  = (int)(((size_t)N * (DFEAT / 4) + TB - 1) / TB);
    const int gScat = (E + 7) / 8;
    const dim3 gGemm((N + 127) / 128, DFEAT / 16);

    // ---- graph-wide normalization (shared by both layers) ----
    hipMemsetAsync(deg, 0, (size_t)N * sizeof(float), stream);
    deg_kernel<<<gE, TB, 0, stream>>>(colp, ew, deg, E);
    dis_kernel<<<gN, TB, 0, stream>>>(deg, dis, N);

    // ---- layer 1: h1 = x@W1 -> out ; z1 -> big ----
    gemm_wmma_f32<<<gGemm, TB, 0, stream>>>(x, W1, out, N);
    selfloop_kernel<<<gVec, TB, 0, stream>>>(out, dis, big, N);
    scatter_kernel<<<gScat, TB, 0, stream>>>(rowp, colp, ew, dis, out, big, E);
    hipMemsetAsync(stats, 0, 2 * DFEAT * sizeof(float), stream);
    stats_kernel<<<gN, TB, 0, stream>>>(big, stats, N);
    bnfin_kernel<<<1, TB, 0, stream>>>(stats, g1, be1, bn, N);
    apply_relu_kernel<<<gVec, TB, 0, stream>>>(big, bn, out, N);   // y1 -> out

    // ---- layer 2: h2 = y1@W2 -> big ; z2 -> out ----
    gemm_wmma_f32<<<gGemm, TB, 0, stream>>>(out, W2, big, N);
    selfloop_kernel<<<gVec, TB, 0, stream>>>(big, dis, out, N);
    scatter_kernel<<<gScat, TB, 0, stream>>>(rowp, colp, ew, dis, big, out, E);
    hipMemsetAsync(stats, 0, 2 * DFEAT * sizeof(float), stream);
    stats_kernel<<<gN, TB, 0, stream>>>(out, stats, N);
    bnfin_kernel<<<1, TB, 0, stream>>>(stats, g2, be2, bn, N);
    apply_final_kernel<<<gVec, TB, 0, stream>>>(out, bn, x, N);    // relu(bn(z2)+x)
}